// DenseCapsule_20358144983616
// MI455X (gfx1250) — compile-verified
//
#include <hip/hip_runtime.h>
#include <hip/hip_bf16.h>

typedef __attribute__((ext_vector_type(2))) float v2f;
typedef __attribute__((ext_vector_type(8))) float v8f;
typedef __attribute__((ext_vector_type(4))) int v4i;
typedef __attribute__((address_space(1))) v4i GLv4i;   // global int4
typedef __attribute__((address_space(3))) v4i LSv4i;   // LDS int4

#define BATCH   64
#define IN_NUM  8192
#define IN_DIM  8
#define OUT_NUM 10
#define OUT_DIM 16
#define YCHUNKS 8

// workspace layout (float offsets)
#define WS_S0 0
#define WS_U  (WS_S0 + BATCH*IN_DIM)
#define WS_V  (WS_U  + BATCH*OUT_NUM*IN_DIM)
#define WS_S1 (WS_V  + BATCH*OUT_NUM)
#define WS_C1 (WS_S1 + BATCH*OUT_NUM*IN_DIM)
#define WS_S2 (WS_C1 + BATCH*OUT_NUM)
#define WS_C2 (WS_S2 + BATCH*OUT_NUM*IN_DIM)
#define WS_END (WS_C2 + BATCH*OUT_NUM)
#define WS_ZERO_BEGIN WS_S1
#define WS_ZERO_COUNT (WS_END - WS_S1)

// ---- CDNA5 async global->LDS path (probe via __has_builtin; sync fallback) --
#if defined(__gfx1250__) && __has_builtin(__builtin_amdgcn_global_load_async_to_lds_b128) && __has_builtin(__builtin_amdgcn_s_wait_asynccnt)
#define USE_ASYNC 1
#else
#define USE_ASYNC 0
#endif

#if __has_builtin(__builtin_amdgcn_s_wait_dscnt)
#define WAIT_DS0() __builtin_amdgcn_s_wait_dscnt(0)
#elif defined(__gfx1250__)
#define WAIT_DS0() asm volatile("s_wait_dscnt 0" ::: "memory")
#else
#define WAIT_DS0()
#endif

// ---------------------------------------------------------------- zero scratch
__global__ void caps_zero(float* __restrict__ p, int n) {
    int i = blockIdx.x * blockDim.x + threadIdx.x;
    if (i < n) p[i] = 0.f;
}

// ------------------------------------------------- pass 0: S0[b,k] = sum_i x
__global__ __launch_bounds__(256) void caps_sum(const float* __restrict__ x,
                                                float* __restrict__ S0) {
    __shared__ float red[256];
    const int b = blockIdx.x, tid = threadIdx.x;
    const int k = tid & 7, ii = tid >> 3;                 // ii in 0..31
    const float* xb = x + (size_t)b * IN_NUM * IN_DIM;
    float acc = 0.f;
    for (int i = ii; i < IN_NUM; i += 32) acc += xb[i * IN_DIM + k];
    red[tid] = acc;
    __syncthreads();
    for (int s = 128; s >= 8; s >>= 1) {
        if (tid < s) red[tid] += red[tid + s];
        __syncthreads();
    }
    if (tid < IN_DIM) S0[b * IN_DIM + tid] = red[tid];
}

// -------------------- squash + build U,V (modes 0/1) or final norms (mode 2)
// mode 0: S = S0[b,k], uniform c=1/10          -> write U,V
// mode 1: S = S1[b,j,k], C = C1[b,j]           -> accumulate U+=,V+=
// mode 2: S = S2, C = C2                       -> out[b,j] = ||squash||_d
__global__ __launch_bounds__(32) void caps_update(
    const float* __restrict__ W, const float* __restrict__ bias,
    const float* __restrict__ S, const float* __restrict__ C,
    float* __restrict__ U, float* __restrict__ V, float* __restrict__ out,
    int mode) {
    __shared__ float outs[OUT_NUM][OUT_DIM];
    const int b = blockIdx.x, t = threadIdx.x;
    if (t < OUT_DIM) {
        float ws[OUT_NUM];
        float nsq = 0.f;
        #pragma unroll
        for (int j = 0; j < OUT_NUM; ++j) {
            float acc = 0.f;
            if (mode == 0) {
                #pragma unroll
                for (int k = 0; k < IN_DIM; ++k)
                    acc += S[b * IN_DIM + k] * W[k * (OUT_NUM * OUT_DIM) + j * OUT_DIM + t];
                acc = (acc + (float)IN_NUM * bias[j * OUT_DIM + t]) * (1.f / OUT_NUM);
            } else {
                #pragma unroll
                for (int k = 0; k < IN_DIM; ++k)
                    acc += S[(b * OUT_NUM + j) * IN_DIM + k] *
                           W[k * (OUT_NUM * OUT_DIM) + j * OUT_DIM + t];
                acc += C[b * OUT_NUM + j] * bias[j * OUT_DIM + t];
            }
            ws[j] = acc;
            nsq += acc * acc;
        }
        // reference squash is over axis 2 == out_num (j), per (b,d)
        const float norm  = sqrtf(nsq);
        const float scale = nsq / (1.f + nsq) / (norm + 1e-8f);
        #pragma unroll
        for (int j = 0; j < OUT_NUM; ++j) outs[j][t] = scale * ws[j];
    }
    __syncthreads();
    if (mode < 2) {
        for (int idx = t; idx < OUT_NUM * IN_DIM; idx += 32) {
            const int j = idx / IN_DIM, k = idx % IN_DIM;
            float u = 0.f;
            #pragma unroll
            for (int d = 0; d < OUT_DIM; ++d)
                u += W[k * (OUT_NUM * OUT_DIM) + j * OUT_DIM + d] * outs[j][d];
            if (mode == 0) U[(b * OUT_NUM + j) * IN_DIM + k] = u;
            else           U[(b * OUT_NUM + j) * IN_DIM + k] += u;
        }
        if (t < OUT_NUM) {
            float v = 0.f;
            #pragma unroll
            for (int d = 0; d < OUT_DIM; ++d) v += bias[t * OUT_DIM + d] * outs[t][d];
            if (mode == 0) V[b * OUT_NUM + t] = v;
            else           V[b * OUT_NUM + t] += v;
        }
    } else {
        if (t < OUT_NUM) {
            float s = 0.f;
            #pragma unroll
            for (int d = 0; d < OUT_DIM; ++d) s += outs[t][d] * outs[t][d];
            out[b * OUT_NUM + t] = sqrtf(s);
        }
    }
}

// --------------- routing pass: logits via WMMA, softmax, accumulate S/C -----
// logits[i,j] = x[b,i,:] . U[b,j,:] + V[b,j]  -> c = softmax_j
// S[b,j,k] += sum_i c*x[b,i,k] ; C[b,j] += sum_i c
__global__ __launch_bounds__(256) void caps_route(
    const float* __restrict__ x, const float* __restrict__ Ug,
    const float* __restrict__ Vg, float* __restrict__ Sout,
    float* __restrict__ Cout) {
    __shared__ float Ul[16][8];
    __shared__ float Vl[16];
    __shared__ float tile[8][2][16][8];   // per-wave double-buffered 16x8 x-tile

    const int b   = blockIdx.x;
    const int yc  = blockIdx.y;
    const int tid = threadIdx.x;

    if (tid < 128) {
        const int j = tid >> 3, k = tid & 7;
        Ul[j][k] = (j < OUT_NUM) ? Ug[(b * OUT_NUM + j) * IN_DIM + k] : 0.f;
    }
    if (tid < 16) Vl[tid] = (tid < OUT_NUM) ? Vg[b * OUT_NUM + tid] : -1e30f;
    __syncthreads();

    const int w    = tid >> 5;
    const int lane = tid & 31;
    const int half = lane >> 4;
    const int j    = lane & 15;

    // B fragments: B[k, j] = U[b][j][k]; 4x16 per K-chunk, 2 VGPRs each
    v2f blo, bhi;
    blo.x = Ul[j][2 * half + 0];  blo.y = Ul[j][2 * half + 1];
    bhi.x = Ul[j][4 + 2 * half];  bhi.y = Ul[j][5 + 2 * half];
    const float vj = Vl[j];

    float accS[8] = {0.f, 0.f, 0.f, 0.f, 0.f, 0.f, 0.f, 0.f};
    float accC = 0.f;

    const int chunks = (IN_NUM / YCHUNKS) / (16 * 8);   // 8

    // per-wave tile fetch: 32 lanes x 16B cover the 16x8 f32 tile
    auto issue = [&](int c, int u) {
        const int i0 = yc * (IN_NUM / YCHUNKS) + (c * 8 + w) * 16;
        const float4* src =
            (const float4*)(x + ((size_t)b * IN_NUM + i0) * IN_DIM) + lane;
        float4* dst = ((float4*)&tile[w][u][0][0]) + lane;
#if USE_ASYNC
        WAIT_DS0();   // WAR: prior ds_loads of this buffer vs async LDS write
        __builtin_amdgcn_global_load_async_to_lds_b128(
            (GLv4i*)src, (LSv4i*)dst, 0, 0);
#else
        *dst = *src;
#endif
    };

    issue(0, 0);
    for (int c = 0; c < chunks; ++c) {
        const int cur = c & 1;
        if (c + 1 < chunks) issue(c + 1, cur ^ 1);
#if USE_ASYNC
        if (c + 1 < chunks) __builtin_amdgcn_s_wait_asynccnt(1);
        else                __builtin_amdgcn_s_wait_asynccnt(0);
        asm volatile("" ::: "memory");   // keep LDS reads below the wait
#endif
        // A fragments: row r = j, K-chunks [0..3] and [4..7]
        const float* myrow = &tile[w][cur][j][0];
        v2f a0, a1;
        a0.x = myrow[2 * half + 0];  a0.y = myrow[2 * half + 1];
        a1.x = myrow[4 + 2 * half];  a1.y = myrow[5 + 2 * half];

        v8f d = {};
        d = __builtin_amdgcn_wmma_f32_16x16x4_f32(false, a0, false, blo,
                                                  (short)0, d, false, false);
        d = __builtin_amdgcn_wmma_f32_16x16x4_f32(false, a1, false, bhi,
                                                  (short)0, d, false, false);

        #pragma unroll
        for (int v = 0; v < 8; ++v) {
            // row m = v + 8*half lives across this 16-lane half (n = j)
            float t = d[v] + vj;
            float mx = t;
            mx = fmaxf(mx, __shfl_xor(mx, 8, 32));
            mx = fmaxf(mx, __shfl_xor(mx, 4, 32));
            mx = fmaxf(mx, __shfl_xor(mx, 2, 32));
            mx = fmaxf(mx, __shfl_xor(mx, 1, 32));
            float e = __expf(t - mx);                // j>=10 -> exp(-huge)=0
            float s = e;
            s += __shfl_xor(s, 8, 32);
            s += __shfl_xor(s, 4, 32);
            s += __shfl_xor(s, 2, 32);
            s += __shfl_xor(s, 1, 32);
            const float cv = e / s;
            accC += cv;
            const float* row = &tile[w][cur][v + 8 * half][0];
            #pragma unroll
            for (int k = 0; k < 8; ++k) accS[k] += cv * row[k];
        }
    }

    // merge the two 16-lane halves (rows 0..7 with rows 8..15, same j)
    accC += __shfl_xor(accC, 16, 32);
    #pragma unroll
    for (int k = 0; k < 8; ++k) accS[k] += __shfl_xor(accS[k], 16, 32);

    if (lane < OUT_NUM) {
        atomicAdd(&Cout[b * OUT_NUM + lane], accC);
        #pragma unroll
        for (int k = 0; k < 8; ++k)
            atomicAdd(&Sout[(b * OUT_NUM + lane) * IN_DIM + k], accS[k]);
    }
}

// ---------------------------------------------------------------------------
extern "C" void kernel_launch(void* const* d_in, const int* in_sizes, int n_in,
                              void* d_out, int out_size, void* d_ws, size_t ws_size,
                              hipStream_t stream) {
    const float* x    = (const float*)d_in[0];
    const float* W    = (const float*)d_in[1];
    const float* bias = (const float*)d_in[2];
    float* ws  = (float*)d_ws;
    float* S0  = ws + WS_S0;
    float* U   = ws + WS_U;
    float* V   = ws + WS_V;
    float* S1  = ws + WS_S1;
    float* C1  = ws + WS_C1;
    float* S2  = ws + WS_S2;
    float* C2  = ws + WS_C2;
    float* out = (float*)d_out;

    caps_zero<<<(WS_ZERO_COUNT + 255) / 256, 256, 0, stream>>>(ws + WS_ZERO_BEGIN,
                                                               WS_ZERO_COUNT);
    caps_sum<<<BATCH, 256, 0, stream>>>(x, S0);
    // iter 0 (uniform c): outputs0 -> U,V
    caps_update<<<BATCH, 32, 0, stream>>>(W, bias, S0, nullptr, U, V, nullptr, 0);
    // iter 1: logits = x.U + V -> softmax -> S1,C1
    caps_route<<<dim3(BATCH, YCHUNKS), 256, 0, stream>>>(x, U, V, S1, C1);
    caps_update<<<BATCH, 32, 0, stream>>>(W, bias, S1, C1, U, V, nullptr, 1);
    // iter 2: accumulated logits -> S2,C2 -> final norms
    caps_route<<<dim3(BATCH, YCHUNKS), 256, 0, stream>>>(x, U, V, S2, C2);
    caps_update<<<BATCH, 32, 0, stream>>>(W, bias, S2, C2, nullptr, nullptr, out, 2);
}